// MultiHeadAttention_11776800325842
// MI455X (gfx1250) — compile-verified
//
#include <hip/hip_runtime.h>

typedef __attribute__((ext_vector_type(2))) float v2f;
typedef __attribute__((ext_vector_type(4))) float v4f;
typedef __attribute__((ext_vector_type(8))) float v8f;

#define S_LEN 2048
#define DMODEL 1024
#define NHEADS 16
#define HDIM 64
#define BATCH 2
#define MROWS (BATCH * S_LEN) /* 4096 */

__device__ __forceinline__ v8f wmma4(v2f a, v2f b, v8f c) {
  // V_WMMA_F32_16X16X4_F32 : D = A(16x4) * B(4x16) + C, fp32 throughout
  return __builtin_amdgcn_wmma_f32_16x16x4_f32(
      false, a, false, b, (short)0, c, false, false);
}

// ---------------------------------------------------------------------------
// Shared GEMM tile core: block = 64(M) x 128(N), 256 threads (8 waves as
// 2(M) x 4(N)); each wave computes a 32x32 block as 2x2 WMMA tiles so every
// pair of A loads + quad of B loads feeds 4 WMMAs (halves LDS bytes/WMMA).
// K is consumed in 32-wide chunks staged in LDS.
// ---------------------------------------------------------------------------
struct GemmAcc { v8f c[2][2]; };

template <typename EpilogueFn>
__device__ __forceinline__ void gemm_64x128(
    const float* __restrict__ X, const float* __restrict__ W,
    int m0, int nb, EpilogueFn epilogue) {
  __shared__ __align__(16) float Xs[64][36];   // pad 36: 16 distinct banks, b128-ok
  __shared__ __align__(16) float Ws[32][136];  // pad 136: halves on disjoint banks

  const int tid  = threadIdx.x;
  const int wave = tid >> 5;
  const int lane = tid & 31;
  const int l15  = lane & 15;
  const int off2 = (lane >> 4) << 1;  // 0 / 2
  const int wave_m = wave >> 2;       // 0..1
  const int wave_n = wave & 3;        // 0..3
  const int mbase = wave_m * 32;      // within block
  const int nbase = wave_n * 32;

  GemmAcc acc;
#pragma unroll
  for (int i = 0; i < 2; ++i)
#pragma unroll
    for (int j = 0; j < 2; ++j)
      acc.c[i][j] = (v8f){0.f,0.f,0.f,0.f,0.f,0.f,0.f,0.f};

  for (int k0 = 0; k0 < DMODEL; k0 += 32) {
    // A tile 64x32 (2048 elems, 8/thread as 2x b128)
#pragma unroll
    for (int it = 0; it < 2; ++it) {
      int e = tid * 4 + it * 1024;
      int r = e >> 5, c = e & 31;
      *(v4f*)&Xs[r][c] = *(const v4f*)&X[(size_t)(m0 + r) * DMODEL + k0 + c];
    }
    // B tile 32x128 (4096 elems, 16/thread as 4x b128)
#pragma unroll
    for (int it = 0; it < 4; ++it) {
      int e = tid * 4 + it * 1024;
      int r = e >> 7, c = e & 127;
      *(v4f*)&Ws[r][c] = *(const v4f*)&W[(size_t)(k0 + r) * DMODEL + nb + c];
    }
    __syncthreads();
#pragma unroll
    for (int kk = 0; kk < 8; ++kk) {
      int kc = kk * 4 + off2;
      v2f a0 = *(const v2f*)&Xs[mbase + l15][kc];
      v2f a1 = *(const v2f*)&Xs[mbase + 16 + l15][kc];
      v2f b0, b1;
      b0[0] = Ws[kc][nbase + l15];
      b0[1] = Ws[kc + 1][nbase + l15];
      b1[0] = Ws[kc][nbase + 16 + l15];
      b1[1] = Ws[kc + 1][nbase + 16 + l15];
      acc.c[0][0] = wmma4(a0, b0, acc.c[0][0]);
      acc.c[0][1] = wmma4(a0, b1, acc.c[0][1]);
      acc.c[1][0] = wmma4(a1, b0, acc.c[1][0]);
      acc.c[1][1] = wmma4(a1, b1, acc.c[1][1]);
    }
    __syncthreads();
  }
  epilogue(acc, m0 + mbase, nb + nbase, l15, (lane >> 4) * 8);
}

// ---------------------------------------------------------------------------
// Kernel 1: fused QKV projection.  grid = (M/64, N/128, 3), block = 256.
// Output layout: [b, h, s, hd]; z==0 (Q) additionally scaled by script weight.
// ---------------------------------------------------------------------------
__global__ __launch_bounds__(256) void qkv_gemm_kernel(
    const float* __restrict__ X,
    const float* __restrict__ Wq, const float* __restrict__ bq,
    const float* __restrict__ Wk, const float* __restrict__ bk,
    const float* __restrict__ Wv, const float* __restrict__ bv,
    const int* __restrict__ script_types, const float* __restrict__ script_w,
    float* __restrict__ qout, float* __restrict__ kout, float* __restrict__ vout) {
  const int z = blockIdx.z;
  const float* W    = (z == 0) ? Wq : (z == 1) ? Wk : Wv;
  const float* bias = (z == 0) ? bq : (z == 1) ? bk : bv;
  float* out        = (z == 0) ? qout : (z == 1) ? kout : vout;
  const int m0 = blockIdx.x * 64;
  const int nb = blockIdx.y * 128;

  gemm_64x128(X, W, m0, nb,
    [&](const GemmAcc& acc, int mrow0, int ncol0, int l15, int half8) {
#pragma unroll
      for (int nt = 0; nt < 2; ++nt) {
        const int ncol = ncol0 + nt * 16 + l15;
        const float bval = bias[ncol];
        const int h = ncol >> 6;
        const int d = ncol & 63;
#pragma unroll
        for (int mt = 0; mt < 2; ++mt) {
#pragma unroll
          for (int j = 0; j < 8; ++j) {
            int r = mrow0 + mt * 16 + j + half8;  // global row = (b, s)
            float val = acc.c[mt][nt][j] + bval;
            if (z == 0) {
              int st = script_types[r];
              val *= script_w[st * NHEADS + h];
            }
            int bi = r >> 11;
            int s  = r & (S_LEN - 1);
            out[((size_t)(bi * NHEADS + h) * S_LEN + s) * HDIM + d] = val;
          }
        }
      }
    });
}

// ---------------------------------------------------------------------------
// Kernel 3: output projection.  grid = (M/64, N/128), block = 256.
// ---------------------------------------------------------------------------
__global__ __launch_bounds__(256) void out_gemm_kernel(
    const float* __restrict__ X, const float* __restrict__ W,
    const float* __restrict__ bias, float* __restrict__ out) {
  const int m0 = blockIdx.x * 64;
  const int nb = blockIdx.y * 128;
  gemm_64x128(X, W, m0, nb,
    [&](const GemmAcc& acc, int mrow0, int ncol0, int l15, int half8) {
#pragma unroll
      for (int nt = 0; nt < 2; ++nt) {
        const int ncol = ncol0 + nt * 16 + l15;
        const float bval = bias[ncol];
#pragma unroll
        for (int mt = 0; mt < 2; ++mt) {
#pragma unroll
          for (int j = 0; j < 8; ++j) {
            int r = mrow0 + mt * 16 + j + half8;
            out[(size_t)r * DMODEL + ncol] = acc.c[mt][nt][j] + bval;
          }
        }
      }
    });
}

// ---------------------------------------------------------------------------
// Kernel 2: attention. grid = (S/16, H, B), block = 256 (8 waves).
// Each block: 16 query rows x 1024 of the 2048 keys. Wave w owns key columns
// [w*256, w*256+128) kept in registers (8 WMMA tiles). A-operand (Q slices)
// hoisted into registers once and reused by all 8 key tiles. Softmax stats
// reduced with wave32 shfl_xor + LDS across waves. probs written from regs.
// ctx = P @ V via per-wave LDS transpose staging, then ds_add_f32 reduction.
// ---------------------------------------------------------------------------
__global__ __launch_bounds__(256) void attn_kernel(
    const float* __restrict__ Q, const float* __restrict__ K,
    const float* __restrict__ V, const int* __restrict__ mask,
    float* __restrict__ probs, float* __restrict__ ctx) {
  __shared__ __align__(16) float Qs[16][68];
  __shared__ __align__(16) float St[8][16][18];
  __shared__ __align__(16) float ctxAcc[16][68];
  __shared__ float smax[8][16];
  __shared__ float ssum[8][16];
  __shared__ float frow[16];

  const int q0   = blockIdx.x * 16;
  const int h    = blockIdx.y;
  const int b    = blockIdx.z;
  const int tid  = threadIdx.x;
  const int wave = tid >> 5;
  const int lane = tid & 31;
  const int l15  = lane & 15;
  const int off2 = (lane >> 4) << 1;
  const int half8 = (lane >> 4) * 8;

  const float* Qb = Q + (size_t)(b * NHEADS + h) * S_LEN * HDIM;
  const float* Kb = K + (size_t)(b * NHEADS + h) * S_LEN * HDIM;
  const float* Vb = V + (size_t)(b * NHEADS + h) * S_LEN * HDIM;

  // init ctx accumulator + load Q block (16 x 64)
  for (int i = tid; i < 16 * 68; i += 256) (&ctxAcc[0][0])[i] = 0.0f;
  {
    int idx = tid * 4;
    int r = idx >> 6, c = idx & 63;
    *(v4f*)&Qs[r][c] = *(const v4f*)&Qb[(size_t)(q0 + r) * HDIM + c];
  }
  __syncthreads();

  // Hoist A operand: same Q slices feed every key tile of this wave.
  v2f areg[16];
#pragma unroll
  for (int kk = 0; kk < 16; ++kk)
    areg[kk] = *(const v2f*)&Qs[l15][kk * 4 + off2];

  // ---- Phase 1: scores = (Q * sw) @ K^T / 8, + mask ----
  v8f p[8];
  const float scl = 0.125f;  // 1/sqrt(64)
#pragma unroll
  for (int t = 0; t < 8; ++t) {
    int n0 = wave * 256 + t * 16;
    v8f acc = {0.f, 0.f, 0.f, 0.f, 0.f, 0.f, 0.f, 0.f};
    const float* Krow = Kb + (size_t)(n0 + l15) * HDIM;
    if (t < 7)  // prefetch next key tile row for this lane
      __builtin_prefetch(Kb + (size_t)(n0 + 16 + l15) * HDIM, 0, 1);
#pragma unroll
    for (int kk = 0; kk < 16; ++kk) {
      v2f bb = *(const v2f*)&Krow[kk * 4 + off2];
      acc = wmma4(areg[kk], bb, acc);
    }
    int mval = mask[b * S_LEN + n0 + l15];  // one key column per lane
#pragma unroll
    for (int j = 0; j < 8; ++j)
      acc[j] = (mval == 0) ? -__builtin_inff() : acc[j] * scl;
    p[t] = acc;
  }

  // ---- Phase 2: softmax row max ----
  {
    float rmax[8];
#pragma unroll
    for (int j = 0; j < 8; ++j) {
      float m = p[0][j];
#pragma unroll
      for (int t = 1; t < 8; ++t) m = fmaxf(m, p[t][j]);
#pragma unroll
      for (int dlt = 1; dlt < 16; dlt <<= 1) m = fmaxf(m, __shfl_xor(m, dlt, 32));
      rmax[j] = m;
    }
    if (l15 == 0) {
#pragma unroll
      for (int j = 0; j < 8; ++j) smax[wave][j + half8] = rmax[j];
    }
  }
  __syncthreads();
  if (tid < 16) {
    float m = smax[0][tid];
    for (int w2 = 1; w2 < 8; ++w2) m = fmaxf(m, smax[w2][tid]);
    frow[tid] = m;
  }
  __syncthreads();

  // ---- Phase 3: exp, row sum, normalize, write probs ----
  {
    float rsum[8];
#pragma unroll
    for (int j = 0; j < 8; ++j) {
      float fm = frow[j + half8];
      float s = 0.f;
#pragma unroll
      for (int t = 0; t < 8; ++t) {
        float e = __expf(p[t][j] - fm);
        p[t][j] = e;
        s += e;
      }
#pragma unroll
      for (int dlt = 1; dlt < 16; dlt <<= 1) s += __shfl_xor(s, dlt, 32);
      rsum[j] = s;
    }
    if (l15 == 0) {
#pragma unroll
      for (int j = 0; j < 8; ++j) ssum[wave][j + half8] = rsum[j];
    }
  }
  __syncthreads();
  if (tid < 16) {
    float s = ssum[0][tid];
    for (int w2 = 1; w2 < 8; ++w2) s += ssum[w2][tid];
    frow[tid] = 1.0f / s;
  }
  __syncthreads();

  float* probsBase = probs + ((size_t)(b * NHEADS + h) * S_LEN + q0) * S_LEN;
#pragma unroll
  for (int j = 0; j < 8; ++j) {
    float inv = frow[j + half8];
    int r = j + half8;
#pragma unroll
    for (int t = 0; t < 8; ++t) {
      float v = p[t][j] * inv;
      p[t][j] = v;
      probsBase[(size_t)r * S_LEN + wave * 256 + t * 16 + l15] = v;
    }
  }

  // ---- Phase 4: ctx partial = P_tile @ V over this wave's key range ----
  v8f cacc[4];
#pragma unroll
  for (int nt = 0; nt < 4; ++nt) cacc[nt] = (v8f){0.f,0.f,0.f,0.f,0.f,0.f,0.f,0.f};
#pragma unroll
  for (int t = 0; t < 8; ++t) {
    // stage 16x16 tile: D-layout registers -> LDS (wave-private, in-order DS)
#pragma unroll
    for (int j = 0; j < 8; ++j) St[wave][j + half8][l15] = p[t][j];
    int kbase = wave * 256 + t * 16;
#pragma unroll
    for (int kk = 0; kk < 4; ++kk) {
      int kc = kk * 4 + off2;
      v2f a = *(const v2f*)&St[wave][l15][kc];
#pragma unroll
      for (int nt = 0; nt < 4; ++nt) {
        v2f bb;
        bb[0] = Vb[(size_t)(kbase + kc) * HDIM + nt * 16 + l15];
        bb[1] = Vb[(size_t)(kbase + kc + 1) * HDIM + nt * 16 + l15];
        cacc[nt] = wmma4(a, bb, cacc[nt]);
      }
    }
  }

  // ---- Phase 5: cross-wave reduce (ds_add_f32) and store ctx [b,s,h,hd] ----
#pragma unroll
  for (int nt = 0; nt < 4; ++nt)
#pragma unroll
    for (int j = 0; j < 8; ++j)
      atomicAdd(&ctxAcc[j + half8][nt * 16 + l15], cacc[nt][j]);
  __syncthreads();
  {
    int idx = tid * 4;
    int r = idx >> 6, c = idx & 63;
    v4f val = *(const v4f*)&ctxAcc[r][c];
    *(v4f*)&ctx[(size_t)(b * S_LEN + q0 + r) * DMODEL + h * HDIM + c] = val;
  }
}

extern "C" void kernel_launch(void* const* d_in, const int* in_sizes, int n_in,
                              void* d_out, int out_size, void* d_ws, size_t ws_size,
                              hipStream_t stream) {
  const float* hidden = (const float*)d_in[0];
  const int*   amask  = (const int*)d_in[1];
  const int*   stypes = (const int*)d_in[2];
  const float* Wq = (const float*)d_in[3];
  const float* bq = (const float*)d_in[4];
  const float* Wk = (const float*)d_in[5];
  const float* bk = (const float*)d_in[6];
  const float* Wv = (const float*)d_in[7];
  const float* bv = (const float*)d_in[8];
  const float* Wo = (const float*)d_in[9];
  const float* bo = (const float*)d_in[10];
  const float* sw = (const float*)d_in[11];

  float* out   = (float*)d_out;
  float* probs = out + (size_t)MROWS * DMODEL;  // outputs: (out, probs) concat

  const size_t QKV_ELEMS = (size_t)MROWS * DMODEL;  // 4,194,304
  float* qbuf = (float*)d_ws;
  float* kbuf = qbuf + QKV_ELEMS;
  float* vbuf = kbuf + QKV_ELEMS;
  float* cbuf = vbuf + QKV_ELEMS;

  qkv_gemm_kernel<<<dim3(MROWS / 64, DMODEL / 128, 3), 256, 0, stream>>>(
      hidden, Wq, bq, Wk, bk, Wv, bv, stypes, sw, qbuf, kbuf, vbuf);
  attn_kernel<<<dim3(S_LEN / 16, NHEADS, BATCH), 256, 0, stream>>>(
      qbuf, kbuf, vbuf, amask, probs, cbuf);
  out_gemm_kernel<<<dim3(MROWS / 64, DMODEL / 128), 256, 0, stream>>>(
      cbuf, Wo, bo, out);
}